// Net_48610439856713
// MI455X (gfx1250) — compile-verified
//
#include <hip/hip_runtime.h>
#include <math.h>

// Problem constants (from reference setup_inputs)
#define BATCH   512
#define IN_F    193
#define L1N     16000
#define L2N     16000
#define L3N     15999
#define GROUP   5333       // L3N / 3
#define NSLICE  32         // o-slices per (btile, group) in the WMMA reduction
#define INV_TAU 0.01f

// Workspace layout (float offsets)
#define OFF_XT   0u                  // 193*512      = 98816
#define OFF_C1   98816u              // 16000*4      = 64000
#define OFF_C2   162816u             // 16000*4      = 64000
#define OFF_C3   226816u             // 15999*4 (pad 64000)
#define OFF_BUFA 290816u             // 16000*512    = 8192000
#define OFF_BUFB 8482816u            // 16000*512    = 8192000
// total 16,674,816 floats (~63.6 MiB)

typedef float v2f __attribute__((ext_vector_type(2)));
typedef float v8f __attribute__((ext_vector_type(8)));

// ---------------------------------------------------------------------------
// Zero the (512 x 3) output (d_out is poisoned by the harness).
// ---------------------------------------------------------------------------
__global__ void zero_out_kernel(float* __restrict__ out) {
    int t = blockIdx.x * blockDim.x + threadIdx.x;
    if (t < BATCH * 3) out[t] = 0.0f;
}

// ---------------------------------------------------------------------------
// Per-neuron softmax over 16 gate logits, collapsed to 4 affine coefficients
// in the basis {1, a, b, ab}.
// ---------------------------------------------------------------------------
__global__ void coef_kernel(const float* __restrict__ w,
                            float* __restrict__ c4, int L) {
    int o = blockIdx.x * blockDim.x + threadIdx.x;
    if (o >= L) return;
    float p[16];
    float m = -1e30f;
    #pragma unroll
    for (int g = 0; g < 16; ++g) { p[g] = w[o * 16 + g]; m = fmaxf(m, p[g]); }
    float s = 0.0f;
    #pragma unroll
    for (int g = 0; g < 16; ++g) { p[g] = __expf(p[g] - m); s += p[g]; }
    float inv = 1.0f / s;
    // Gate -> {1,a,b,ab} coefficient table (see analysis)
    float c0 = (p[8] + p[9] + p[10] + p[11] + p[12] + p[13] + p[14] + p[15]) * inv;
    float c1 = (p[2] + p[3] + p[6] + p[7] - p[8] - p[9] - p[12] - p[13]) * inv;
    float c2 = (p[4] + p[5] + p[6] + p[7] - p[8] - p[9] - p[10] - p[11]) * inv;
    float c3 = (p[1] - p[2] - p[4] - 2.0f * p[6] - p[7] + p[8] + 2.0f * p[9]
                + p[11] + p[13] - p[14]) * inv;
    float4 c = make_float4(c0, c1, c2, c3);
    ((float4*)c4)[o] = c;
}

// ---------------------------------------------------------------------------
// Transpose x: (512 x 193) row-major -> xT: (193 x 512) feature-major.
// ---------------------------------------------------------------------------
__global__ void transpose_x_kernel(const float* __restrict__ x,
                                   float* __restrict__ xT) {
    int t = blockIdx.x * blockDim.x + threadIdx.x;
    if (t >= IN_F * BATCH) return;
    int c = t / BATCH, b = t % BATCH;
    xT[t] = x[b * IN_F + c];  // coalesced write; tiny strided read
}

// ---------------------------------------------------------------------------
// One logic layer, feature-major both sides. One block per neuron o;
// 128 threads x float4 cover the 512-wide batch. Indices & coefficients are
// block-uniform -> scalar loads. Gathers and stores are fully coalesced.
// ---------------------------------------------------------------------------
__global__ void __launch_bounds__(128)
logic_layer_kernel(const float* __restrict__ inT,
                   const float* __restrict__ c4,
                   const int* __restrict__ ia,
                   const int* __restrict__ ib,
                   float* __restrict__ outT) {
    const int o = blockIdx.x;
    const int t = threadIdx.x;                       // 0..127
    const float4 c  = ((const float4*)c4)[o];
    const int A = ia[o];
    const int Bx = ib[o];
    const float4 av = ((const float4*)(inT + (size_t)A  * BATCH))[t];
    const float4 bv = ((const float4*)(inT + (size_t)Bx * BATCH))[t];
    float4 r;
    r.x = fmaf(av.x * bv.x, c.w, fmaf(bv.x, c.z, fmaf(av.x, c.y, c.x)));
    r.y = fmaf(av.y * bv.y, c.w, fmaf(bv.y, c.z, fmaf(av.y, c.y, c.x)));
    r.z = fmaf(av.z * bv.z, c.w, fmaf(bv.z, c.z, fmaf(av.z, c.y, c.x)));
    r.w = fmaf(av.w * bv.w, c.w, fmaf(bv.w, c.z, fmaf(av.w, c.y, c.x)));
    ((float4*)(outT + (size_t)o * BATCH))[t] = r;
}

// ---------------------------------------------------------------------------
// Group-sum reduction via V_WMMA_F32_16X16X4_F32:
//   out[b,k] = (1/tau) * sum_{o in group k} h3T[o][b]
// A (16x4) tile: lane L<16 holds {K=0,K=1}, lane L>=16 holds {K=2,K=3}
// (per CDNA5 ISA 32-bit A layout). B (4x16) = all ones, so every column of D
// accumulates the row-sum. Main loop is branchless (only full 4-neuron
// chunks); the single straddling chunk per group (GROUP % 4 == 1) is handled
// in a branchless epilogue via index-clamp + multiply mask (v_cndmask, no
// exec divergence, no out-of-group reads). Partials across NSLICE o-slices
// are combined with global_atomic_add_f32.
// Grid: (32 b-tiles, 3 groups, NSLICE). Block: 32 lanes (one wave).
// ---------------------------------------------------------------------------
__global__ void __launch_bounds__(32)
reduce_wmma_kernel(const float* __restrict__ h3T, float* __restrict__ out) {
    const int btile = blockIdx.x;        // 0..31
    const int k     = blockIdx.y;        // 0..2
    const int slice = blockIdx.z;        // 0..NSLICE-1
    const int lane  = threadIdx.x;       // 0..31
    const int b0    = btile * 16;
    const int o_beg = k * GROUP;
    const int o_end = o_beg + GROUP;

    const int laneM = lane & 15;         // A-matrix M row
    const int hi    = lane >> 4;         // K-half: 0 -> K0/K1, 1 -> K2/K3

    v8f acc = {};
    v2f bmat;
    bmat[0] = 1.0f;
    bmat[1] = 1.0f;

    // Main loop: only chunks fully inside the group -> unconditional loads.
    const size_t step = (size_t)NSLICE * 4 * BATCH;
    const float* p = h3T + (size_t)(o_beg + slice * 4 + 2 * hi) * BATCH
                          + b0 + laneM;
    int o = o_beg + slice * 4;
    #pragma unroll 2
    for (; o + 4 <= o_end; o += NSLICE * 4, p += step) {
        v2f amat;
        amat[0] = p[0];
        amat[1] = p[BATCH];
        acc = __builtin_amdgcn_wmma_f32_16x16x4_f32(
                  false, amat, false, bmat, (short)0, acc, false, false);
    }

    // Epilogue: at most one partial chunk (owned by exactly one slice).
    if (o < o_end) {
        const int o0 = o + 2 * hi;
        const int o1 = o0 + 1;
        const int i0 = (o0 < o_end) ? o0 : (o_end - 1);   // clamped, in-group
        const int i1 = (o1 < o_end) ? o1 : (o_end - 1);
        const float m0 = (o0 < o_end) ? 1.0f : 0.0f;
        const float m1 = (o1 < o_end) ? 1.0f : 0.0f;
        v2f amat;
        amat[0] = h3T[(size_t)i0 * BATCH + b0 + laneM] * m0;
        amat[1] = h3T[(size_t)i1 * BATCH + b0 + laneM] * m1;
        acc = __builtin_amdgcn_wmma_f32_16x16x4_f32(
                  false, amat, false, bmat, (short)0, acc, false, false);
    }

    // D layout: VGPR r -> M=r (lanes 0-15) / M=8+r (lanes 16-31), replicated
    // across N. Lane 0 carries rows 0..7, lane 16 carries rows 8..15.
    if (lane == 0 || lane == 16) {
        const int mbase = (lane == 0) ? 0 : 8;
        #pragma unroll
        for (int r = 0; r < 8; ++r) {
            atomicAdd(&out[(b0 + mbase + r) * 3 + k], acc[r] * INV_TAU);
        }
    }
}

// ---------------------------------------------------------------------------
extern "C" void kernel_launch(void* const* d_in, const int* in_sizes, int n_in,
                              void* d_out, int out_size, void* d_ws, size_t ws_size,
                              hipStream_t stream) {
    const float* x     = (const float*)d_in[0];
    const float* w1    = (const float*)d_in[1];
    const float* w2    = (const float*)d_in[2];
    const float* w3    = (const float*)d_in[3];
    const int*   idx1a = (const int*)d_in[4];
    const int*   idx1b = (const int*)d_in[5];
    const int*   idx2a = (const int*)d_in[6];
    const int*   idx2b = (const int*)d_in[7];
    const int*   idx3a = (const int*)d_in[8];
    const int*   idx3b = (const int*)d_in[9];
    float* out = (float*)d_out;

    float* ws   = (float*)d_ws;
    float* xT   = ws + OFF_XT;
    float* c1   = ws + OFF_C1;
    float* c2   = ws + OFF_C2;
    float* c3   = ws + OFF_C3;
    float* bufA = ws + OFF_BUFA;   // h1T, later reused as h3T
    float* bufB = ws + OFF_BUFB;   // h2T

    // 0) zero the output (it is poisoned before timing)
    zero_out_kernel<<<(BATCH * 3 + 255) / 256, 256, 0, stream>>>(out);

    // 1) collapse softmax(w) -> 4 affine coefficients per neuron
    coef_kernel<<<(L1N + 255) / 256, 256, 0, stream>>>(w1, c1, L1N);
    coef_kernel<<<(L2N + 255) / 256, 256, 0, stream>>>(w2, c2, L2N);
    coef_kernel<<<(L3N + 255) / 256, 256, 0, stream>>>(w3, c3, L3N);

    // 2) transpose x to feature-major so all gathers coalesce
    transpose_x_kernel<<<(IN_F * BATCH + 255) / 256, 256, 0, stream>>>(x, xT);

    // 3) three logic layers, feature-major ping-pong
    logic_layer_kernel<<<L1N, 128, 0, stream>>>(xT,   c1, idx1a, idx1b, bufA);
    logic_layer_kernel<<<L2N, 128, 0, stream>>>(bufA, c2, idx2a, idx2b, bufB);
    logic_layer_kernel<<<L3N, 128, 0, stream>>>(bufB, c3, idx3a, idx3b, bufA);

    // 4) WMMA ones-matmul group reduction: (512 x 15999) -> (512 x 3)
    dim3 rgrid(BATCH / 16, 3, NSLICE);
    reduce_wmma_kernel<<<rgrid, 32, 0, stream>>>(bufA, out);
}